// PointNetSetAbstraction_68650757259520
// MI455X (gfx1250) — compile-verified
//
#include <hip/hip_runtime.h>
#include <math.h>

// Problem constants (match reference)
#define B_  8
#define N_  4096
#define D_  64
#define S_  1024   // NPOINT
#define K_  32     // NSAMPLE

typedef __attribute__((ext_vector_type(2))) float v2f;
typedef __attribute__((ext_vector_type(4))) float v4f;
typedef __attribute__((ext_vector_type(8))) float v8f;

// ---------------------------------------------------------------------------
// Kernel 1: Farthest Point Sampling. One workgroup per batch, 1024 threads
// (32 waves). xyz kept in LDS as float4 {x,y,z,0} so the serial loop does one
// ds_load_b128 per point; per-thread running min-distance in registers;
// block argmax = wave32 butterfly + 32-partial LDS reduce.
// Also stages the WMMA A-matrix rows (-2x,-2y,-2z,1) and |c|^2 into workspace
// and writes new_xyz (B,3,S) directly to the output.
// ---------------------------------------------------------------------------
__global__ __launch_bounds__(1024) void fps_kernel(
    const float* __restrict__ xyz,           // (B,3,N)
    const int*   __restrict__ farthest_init, // (B,)
    float* __restrict__ out_newxyz,          // (B,3,S)
    float* __restrict__ wsA,                 // (B,S,4)  rows (-2x,-2y,-2z,1)
    float* __restrict__ wsC2)                // (B,S)    |c|^2
{
    __shared__ v4f   p4[N_];      // {x,y,z,0}  64 KB
    __shared__ float redV[32];
    __shared__ int   redI[32];
    __shared__ int   sFar;

    const int b    = blockIdx.x;
    const int tid  = threadIdx.x;
    const int lane = tid & 31;
    const int wid  = tid >> 5;

    const float* xb = xyz + (size_t)b * 3 * N_;
    for (int i = tid; i < N_; i += 1024) {
        v4f v;
        v.x = xb[i];
        v.y = xb[N_ + i];
        v.z = xb[2 * N_ + i];
        v.w = 0.f;
        p4[i] = v;
    }
    float dist[4];
#pragma unroll
    for (int j = 0; j < 4; ++j) dist[j] = 1e10f;
    if (tid == 0) sFar = farthest_init[b];
    __syncthreads();

    for (int it = 0; it < S_; ++it) {
        const v4f c = p4[sFar];
        if (tid == 0) {
            // reference records the incoming `far` each step
            out_newxyz[(size_t)b * 3 * S_ + 0 * S_ + it] = c.x;
            out_newxyz[(size_t)b * 3 * S_ + 1 * S_ + it] = c.y;
            out_newxyz[(size_t)b * 3 * S_ + 2 * S_ + it] = c.z;
            float* A = wsA + ((size_t)b * S_ + it) * 4;
            A[0] = -2.f * c.x; A[1] = -2.f * c.y; A[2] = -2.f * c.z; A[3] = 1.f;
            wsC2[(size_t)b * S_ + it] = c.x * c.x + c.y * c.y + c.z * c.z;
        }
        // local update + argmax (first-index tie-break, like jnp.argmax)
        float bv = -1.f; int bi = 0x7fffffff;
#pragma unroll
        for (int j = 0; j < 4; ++j) {
            const int i = tid + 1024 * j;
            const v4f p = p4[i];
            const float dx = p.x - c.x, dy = p.y - c.y, dz = p.z - c.z;
            const float d = dx * dx + dy * dy + dz * dz;
            const float nd = fminf(dist[j], d);
            dist[j] = nd;
            if (nd > bv || (nd == bv && i < bi)) { bv = nd; bi = i; }
        }
#pragma unroll
        for (int off = 16; off >= 1; off >>= 1) {
            const float ov = __shfl_xor(bv, off, 32);
            const int   oi = __shfl_xor(bi, off, 32);
            if (ov > bv || (ov == bv && oi < bi)) { bv = ov; bi = oi; }
        }
        if (lane == 0) { redV[wid] = bv; redI[wid] = bi; }
        __syncthreads();
        if (wid == 0) {
            bv = redV[lane]; bi = redI[lane];
#pragma unroll
            for (int off = 16; off >= 1; off >>= 1) {
                const float ov = __shfl_xor(bv, off, 32);
                const int   oi = __shfl_xor(bi, off, 32);
                if (ov > bv || (ov == bv && oi < bi)) { bv = ov; bi = oi; }
            }
            if (lane == 0) sFar = bi;
        }
        __syncthreads();
    }
}

// ---------------------------------------------------------------------------
// Kernel 2: per block = 1 batch x 16 centroids.
// Phase A: WMMA f32 16x16x4 distance tiles -> 16 x 4096 tile in LDS
//          (pitch 4100 floats; ~262 KB of the 320 KB WGP LDS). Fragment
//          construction is branchless (cndmask, not exec branches).
// Phase B: 32 argmin passes per row, vectorized ds_load_b128 scans
//          (4 contiguous floats/lane, conflict-free), wave32 butterfly,
//          exact (dist,idx) lexicographic tie-break.
// Phase C: gather 32 neighbors, channel-max over 67 dims, write feat (B,67,S).
// ---------------------------------------------------------------------------
#define TPITCH 4100

__global__ __launch_bounds__(256) void group_kernel(
    const float* __restrict__ xyz,   // (B,3,N)
    const float* __restrict__ pts,   // (B,D,N)
    const float* __restrict__ wsA,   // (B,S,4)
    const float* __restrict__ wsC2,  // (B,S)
    float* __restrict__ out_feat)    // (B,67,S)
{
    __shared__ float DT[16 * TPITCH];
    __shared__ int   selIdx[16][K_];

    const int b    = blockIdx.y;
    const int s0   = blockIdx.x * 16;
    const int tid  = threadIdx.x;
    const int lane = tid & 31;
    const int wid  = tid >> 5;   // 0..7
    const bool hi  = lane >= 16;

    const float* xb = xyz + (size_t)b * 3 * N_;

    // ---- Phase A: WMMA distance tiles ----------------------------------
    // A-matrix 16x4 f32 layout (wave32): lanes 0-15 hold M=0..15 K={0,1},
    // lanes 16-31 hold K={2,3}.  A row m = (-2x,-2y,-2z,1).  Branchless.
    const int  mcol = lane & 15;
    const v4f  A4   = *(const v4f*)(wsA + ((size_t)b * S_ + s0 + mcol) * 4);
    v2f a;
    a.x = hi ? A4.z : A4.x;
    a.y = hi ? A4.w : A4.y;

    // Per-lane row biases |c_m|^2 for the 8 output rows (loop-invariant).
    const int mbase = hi ? 8 : 0;
    float cmr[8];
#pragma unroll
    for (int r = 0; r < 8; ++r)
        cmr[r] = wsC2[(size_t)b * S_ + s0 + mbase + r];

    // B-matrix 4x16 f32: lane n holds column n; lanes 0-15 rows K={0,1},
    // lanes 16-31 rows K={2,3}.  B col n = (x, y, z, |p|^2).  Branchless.
    for (int t = 0; t < 32; ++t) {                 // wave wid owns 512 cols
        const int n = wid * 512 + t * 16 + mcol;
        const float x = xb[n], y = xb[N_ + n], z = xb[2 * N_ + n];
        const float pn2 = x * x + y * y + z * z;
        v2f bb;
        bb.x = hi ? z : x;
        bb.y = hi ? pn2 : y;

        v8f c = {};
        // D = A*B + 0 :  D[m][n] = |p_n|^2 - 2 c_m . p_n
        c = __builtin_amdgcn_wmma_f32_16x16x4_f32(
                /*neg_a=*/false, a, /*neg_b=*/false, bb,
                /*c_mod=*/(short)0, c, /*reuse_a=*/false, /*reuse_b=*/false);

        // C/D layout: VGPR r -> M = r + 8*(lane>=16), N = lane & 15
#pragma unroll
        for (int r = 0; r < 8; ++r)
            DT[(mbase + r) * TPITCH + n] = c[r] + cmr[r];
    }
    __syncthreads();

    // ---- Phase B: top-32 smallest per row ------------------------------
    for (int rr = 0; rr < 2; ++rr) {
        const int m = wid + rr * 8;
        float* row = &DT[m * TPITCH];
        for (int k = 0; k < K_; ++k) {
            float bv = 3.4e38f; int bi = 0x7fffffff;
            // 4 contiguous floats per lane -> ds_load_b128, conflict-free
            for (int jj = 0; jj < N_ / 128; ++jj) {
                const int j0 = jj * 128 + lane * 4;
                const v4f dv = *(const v4f*)&row[j0];
#pragma unroll
                for (int e = 0; e < 4; ++e) {
                    const float d = dv[e];
                    const int   j = j0 + e;
                    if (d < bv || (d == bv && j < bi)) { bv = d; bi = j; }
                }
            }
#pragma unroll
            for (int off = 16; off >= 1; off >>= 1) {
                const float ov = __shfl_xor(bv, off, 32);
                const int   oi = __shfl_xor(bi, off, 32);
                if (ov < bv || (ov == bv && oi < bi)) { bv = ov; bi = oi; }
            }
            if (lane == 0) { selIdx[m][k] = bi; row[bi] = 3.4e38f; }
            __syncthreads();   // uniform across the block (2*32 iterations)
        }
    }
    __syncthreads();

    // ---- Phase C: gather + channel max ---------------------------------
    const float* pb = pts + (size_t)b * D_ * N_;
    for (int t = tid; t < 16 * 67; t += 256) {
        const int m = t / 67;
        const int d = t % 67;
        const float* src = (d < 3) ? (xb + (size_t)d * N_)
                                   : (pb + (size_t)(d - 3) * N_);
        float mx = -3.4e38f;
#pragma unroll 4
        for (int k = 0; k < K_; ++k) mx = fmaxf(mx, src[selIdx[m][k]]);
        out_feat[(size_t)b * 67 * S_ + (size_t)d * S_ + (s0 + m)] = mx;
    }
}

// ---------------------------------------------------------------------------
extern "C" void kernel_launch(void* const* d_in, const int* in_sizes, int n_in,
                              void* d_out, int out_size, void* d_ws, size_t ws_size,
                              hipStream_t stream) {
    const float* xyz    = (const float*)d_in[0];   // (8,3,4096)
    const float* points = (const float*)d_in[1];   // (8,64,4096)
    const int*   finit  = (const int*)d_in[2];     // (8,)

    float* out        = (float*)d_out;
    float* out_newxyz = out;                       // (8,3,1024)
    float* out_feat   = out + (size_t)B_ * 3 * S_; // (8,67,1024)

    float* wsA  = (float*)d_ws;                    // B*S*4 floats
    float* wsC2 = wsA + (size_t)B_ * S_ * 4;       // B*S floats  (~160 KB total)

    fps_kernel<<<B_, 1024, 0, stream>>>(xyz, finit, out_newxyz, wsA, wsC2);
    group_kernel<<<dim3(S_ / 16, B_), 256, 0, stream>>>(xyz, points, wsA, wsC2, out_feat);
}